// PyramidROIAlign_58823872086113
// MI455X (gfx1250) — compile-verified
//
#include <hip/hip_runtime.h>
#include <hip/hip_bf16.h>

// PyramidROIAlign for MI455X (gfx1250, wave32).
// Memory-bound kernel: ~100 MB f32 output writes dominate (≈4.3 us at 23.3 TB/s);
// gathered reads stay L2-resident (pyramid = 178 MB < 192 MB L2).
// Separable bilinear: stage 1 (x-interp) in VALU, stage 2 (y-interp) as a
// 7x14 GEMM per 16-channel tile via chained V_WMMA_F32_16X16X4_F32.

typedef __attribute__((ext_vector_type(2))) float v2f;
typedef __attribute__((ext_vector_type(8))) float v8f;

#define PH 7
#define PW 7
#define CCH 256   // channels
#define CT 16     // channels per WMMA tile (N dimension)

__global__ __launch_bounds__(128)
void roi_align_wmma_kernel(const float* __restrict__ boxes,
                           const float* __restrict__ meta,
                           const float* __restrict__ f2,
                           const float* __restrict__ f3,
                           const float* __restrict__ f4,
                           const float* __restrict__ f5,
                           float* __restrict__ out,
                           int N)
{
    // Per-wave private slab of x-interpolated values: U[j(14 rows)][px(7)][ch(16)]
    __shared__ float U[4][14][PW][CT];

    const int box  = blockIdx.x;          // b*N + n
    const int lane = threadIdx.x & 31;
    const int wave = threadIdx.x >> 5;
    const int b    = box / N;

    // ---- box + level (wave-uniform) ------------------------------------
    const float y1 = boxes[(size_t)box * 4 + 0];
    const float x1 = boxes[(size_t)box * 4 + 1];
    const float y2 = boxes[(size_t)box * 4 + 2];
    const float x2 = boxes[(size_t)box * 4 + 3];

    const float area = meta[4] * meta[5];                 // image_meta[0,4]*[0,5]
    const float hh = y2 - y1, wwd = x2 - x1;
    const float lvl = log2f(sqrtf(hh * wwd) / (224.0f / sqrtf(area)));
    float rl = 4.0f + rintf(lvl);                         // rintf == round-half-even
    rl = fminf(fmaxf(rl, 2.0f), 5.0f);
    const int level = (int)rl;                            // 2..5

    const float* fm;
    int H;
    if (level == 2)      { fm = f2; H = 256; }
    else if (level == 3) { fm = f3; H = 128; }
    else if (level == 4) { fm = f4; H = 64;  }
    else                 { fm = f5; H = 32;  }
    const int   W    = H;
    const float Hm1  = (float)(H - 1), Wm1 = (float)(W - 1);
    const float sy   = (y2 - y1) * Hm1 / 6.0f;            // (ph-1)=6
    const float sx   = (x2 - x1) * Wm1 / 6.0f;
    const float ybase = y1 * Hm1;
    const float xbase = x1 * Wm1;
    const float* fmb = fm + (size_t)b * H * W * CCH;

    // ---- per-lane y-interp weight for stage-2 A fragment ----------------
    // D/A rows: M = lane%16 (pooled row py); rows 7..15 are padding.
    const int   py  = lane & 15;
    const float ysl = ybase + (float)py * sy;
    const float fy  = ysl - floorf(ysl);
    const int   h16 = lane >> 4;                          // half-wave id

    for (int it = 0; it < 4; ++it) {
        const int ct = wave + 4 * it;                     // channel tile 0..15
        const int ch = ct * CT + (lane & 15);

        // ---- stage 1: x interpolation (VALU), U[j][px][ch] --------------
        // j even -> row y0[j/2] (top), j odd -> row y1[j/2] (bottom)
        for (int p2 = 0; p2 < 98; p2 += 2) {
            const int p  = p2 + h16;                      // 0..97
            const int j  = p / PW;
            const int px = p % PW;

            const int   jy  = j >> 1;
            const float ysv = ybase + (float)jy * sy;
            const float y0f = floorf(ysv);
            int y0 = (int)y0f; y0 = min(max(y0, 0), H - 1);
            const int yi = (j & 1) ? min(y0 + 1, H - 1) : y0;

            const float xsv = xbase + (float)px * sx;
            const float x0f = floorf(xsv);
            int x0 = (int)x0f; x0 = min(max(x0, 0), W - 1);
            const int   x1i = min(x0 + 1, W - 1);
            const float fx  = xsv - x0f;

            const float* rp = fmb + (size_t)yi * W * CCH;
            const float tl = rp[(size_t)x0  * CCH + ch];
            const float tr = rp[(size_t)x1i * CCH + ch];
            U[wave][j][px][lane & 15] = tl + (tr - tl) * fx;  // matches reference
        }
        __syncthreads();

        // ---- stage 2: y interpolation via WMMA f32 16x16x4 ---------------
        // out[py,ch] = sum_j Wy[py,j] * U[j,px,ch]; Wy nonzero at j=2py (1-fy)
        // and j=2py+1 (fy). K=14 padded to 16 -> 4 chained K=4 WMMAs.
        for (int px = 0; px < PW; ++px) {
            v8f acc = {0.f, 0.f, 0.f, 0.f, 0.f, 0.f, 0.f, 0.f};
#pragma unroll
            for (int k = 0; k < 4; ++k) {
                v2f a, bbv;
#pragma unroll
                for (int v = 0; v < 2; ++v) {
                    // 32-bit A/B fragment layout: local K = v + 2*h16
                    const int j = 4 * k + v + 2 * h16;    // global K index
                    const float wgt = (j == 2 * py)     ? (1.0f - fy)
                                    : (j == 2 * py + 1) ? fy
                                                        : 0.0f;
                    const float uv = (j < 14) ? U[wave][j][px][lane & 15] : 0.0f;
                    a[v]   = wgt;                         // A[M=py, K=j]
                    bbv[v] = uv;                          // B[K=j, N=ch]
                }
                acc = __builtin_amdgcn_wmma_f32_16x16x4_f32(
                        false, a, false, bbv, (short)0, acc, false, false);
            }
            // D rows M=0..6 live in acc[0..6] on lanes 0-15 (N = lane).
            if (lane < 16) {
                float* op = out + ((size_t)box * PH * PW + px) * CCH + ct * CT + lane;
#pragma unroll
                for (int r = 0; r < PH; ++r) {
                    op[(size_t)r * PW * CCH] = acc[r];
                }
            }
        }
        __syncthreads();
    }
}

extern "C" void kernel_launch(void* const* d_in, const int* in_sizes, int n_in,
                              void* d_out, int out_size, void* d_ws, size_t ws_size,
                              hipStream_t stream) {
    const float* boxes = (const float*)d_in[0];   // (B,N,4) f32
    const float* meta  = (const float*)d_in[1];   // (B,14)  f32
    const float* f2    = (const float*)d_in[2];   // (B,256,256,256)
    const float* f3    = (const float*)d_in[3];   // (B,128,128,256)
    const float* f4    = (const float*)d_in[4];   // (B,64,64,256)
    const float* f5    = (const float*)d_in[5];   // (B,32,32,256)
    float* out         = (float*)d_out;           // (B,N,7,7,256)

    const long f2sz = (long)in_sizes[2];
    int B = (int)(f2sz / (256L * 256L * 256L));
    if (B < 1) B = 1;
    const int N = in_sizes[0] / (4 * B);
    const int nbox = B * N;

    roi_align_wmma_kernel<<<dim3(nbox), dim3(128), 0, stream>>>(
        boxes, meta, f2, f3, f4, f5, out, N);
}